// TemporalAttention5_55138790146546
// MI455X (gfx1250) — compile-verified
//
#include <hip/hip_runtime.h>
#include <hip/hip_bf16.h>
#include <math.h>

typedef __attribute__((ext_vector_type(2))) float v2f;
typedef __attribute__((ext_vector_type(8))) float v8f;

#define T_   512
#define B_   16
#define D_   512
#define G3   1536          // 3*D
#define WIN  12
#define NSEQ 8192          // B_*T_
#define PADROW 8192        // extra gi row holding b_ih (zero-input sequences)
#define HSTRIDE 516        // LDS row stride in floats (conflict-free for b64 reads)

__device__ __forceinline__ v8f wmma4(v2f a, v2f b, v8f c) {
  // D = A(16x4 f32) * B(4x16 f32) + C(16x16 f32)
  return __builtin_amdgcn_wmma_f32_16x16x4_f32(false, a, false, b, (short)0, c, false, false);
}
__device__ __forceinline__ float sigmoidf_(float x) { return 1.f / (1.f + expf(-x)); }

// ---------------------------------------------------------------------------
// Kernel 1: pack W_ih / W_hh into K-major 4-wide layout:
//   W_P[(k>>2)*G3*4 + n*4 + (k&3)] = W[n*D_ + k]
// and write the gi pad row (= b_ih) for zero-input sequences.
// ---------------------------------------------------------------------------
__global__ void prep_kernel(const float* __restrict__ Wih, const float* __restrict__ Whh,
                            const float* __restrict__ bih,
                            float* __restrict__ WihP, float* __restrict__ WhhP,
                            float* __restrict__ gi) {
  int idx = blockIdx.x * blockDim.x + threadIdx.x;
  const int n = G3 * D_;
  if (idx < 2 * n) {
    const float* src = (idx < n) ? Wih : Whh;
    float*       dst = (idx < n) ? WihP : WhhP;
    int i    = (idx < n) ? idx : idx - n;
    int nrow = i / D_;   // 0..1535   (consecutive idx -> consecutive k: coalesced reads)
    int k    = i % D_;
    dst[(k >> 2) * (G3 * 4) + nrow * 4 + (k & 3)] = src[nrow * D_ + k];
  } else if (idx < 2 * n + G3) {
    int j = idx - 2 * n;
    gi[(size_t)PADROW * G3 + j] = bih[j];
  }
}

// ---------------------------------------------------------------------------
// Kernel 2: banded top-12 neighbor indices per (b,t), sorted ascending.
// One wave per row; lane = band candidate; 1/sqrt(d) scaling is monotonic so skipped.
// ---------------------------------------------------------------------------
__global__ void topk_kernel(const float* __restrict__ x, int* __restrict__ ids) {
  const int bt = blockIdx.x;
  const int t  = bt & (T_ - 1);
  const int lane = threadIdx.x;
  int s0 = t - 11; if (s0 < 0) s0 = 0;
  int s1 = t + 11; if (s1 > T_ - 1) s1 = T_ - 1;
  const int nc = s1 - s0 + 1;     // always >= 12
  __shared__ float sc[32];
  if (lane < nc) {
    const float* xt = x + (size_t)bt * D_;
    const float* xs = x + (size_t)(bt - t + s0 + lane) * D_;
    float acc = 0.f;
    for (int k = 0; k < D_; k += 4) {
      float4 a = *(const float4*)(xt + k);
      float4 c = *(const float4*)(xs + k);
      acc += a.x * c.x + a.y * c.y + a.z * c.z + a.w * c.w;
    }
    sc[lane] = acc;
  }
  __syncthreads();
  if (lane == 0) {
    unsigned used = 0u;
    int sel[WIN];
    for (int i = 0; i < WIN; ++i) {
      int best = -1; float bv = 0.f;
      for (int c = 0; c < nc; ++c) {
        if (used & (1u << c)) continue;
        if (best < 0 || sc[c] > bv) { bv = sc[c]; best = c; }   // ties -> lower index
      }
      used |= (1u << best);
      sel[i] = s0 + best;
    }
    for (int i = 1; i < WIN; ++i) {           // insertion sort ascending
      int v = sel[i]; int k = i - 1;
      while (k >= 0 && sel[k] > v) { sel[k + 1] = sel[k]; --k; }
      sel[k + 1] = v;
    }
    for (int i = 0; i < WIN; ++i) ids[bt * WIN + i] = sel[i];
  }
}

// ---------------------------------------------------------------------------
// Kernel 3: dedup by (mean, unbiased std) key, stable compaction.
// Emits widx[seq*12+s] = gi row to gather at step s (PADROW for padding),
// and feat_len[b] (as float) into the output tail.
// One 512-thread block per batch.
// ---------------------------------------------------------------------------
__global__ void dedup_kernel(const int* __restrict__ ids, int* __restrict__ widx,
                             float* __restrict__ out_flen) {
  const int b = blockIdx.x;
  const int t = threadIdx.x;
  __shared__ float km[T_], ks[T_];
  __shared__ int keep[T_];
  __shared__ int kept_src[T_];
  __shared__ int flen;
  const int* row = ids + (size_t)(b * T_ + t) * WIN;
  int v[WIN]; int sum = 0;
  for (int i = 0; i < WIN; ++i) { v[i] = row[i]; sum += v[i]; }
  float m = (float)sum / 12.f;
  float sq = 0.f;
  for (int i = 0; i < WIN; ++i) { float d = (float)v[i] - m; sq += d * d; }
  float s = sqrtf(sq / 11.f);     // ddof=1
  km[t] = m; ks[t] = s;
  __syncthreads();
  int first = t;
  for (int u = 0; u < t; ++u) {
    if (km[u] == m && ks[u] == s) { first = u; break; }
  }
  keep[t] = (first == t) ? 1 : 0;
  __syncthreads();
  if (t == 0) {
    int c = 0;
    for (int u = 0; u < T_; ++u) if (keep[u]) kept_src[c++] = u;
    flen = c;
    out_flen[b] = (float)c;
  }
  __syncthreads();
  const size_t base = (size_t)(b * T_ + t) * WIN;
  if (t < flen) {
    const int src = kept_src[t];
    const int* r2 = ids + (size_t)(b * T_ + src) * WIN;
    for (int i = 0; i < WIN; ++i) widx[base + i] = b * T_ + r2[i];
  } else {
    for (int i = 0; i < WIN; ++i) widx[base + i] = PADROW;
  }
}

// ---------------------------------------------------------------------------
// Kernel 4: gi_all[m] = x[m] @ W_ih^T + b_ih  for m in [0, 8192)
// 32-row M tile in LDS, 16 waves; each wave: 12 WMMA N-tiles (16x16), K=512.
// ---------------------------------------------------------------------------
__global__ __launch_bounds__(512) void gi_gemm_kernel(const float* __restrict__ x,
                                                      const float* __restrict__ WihP,
                                                      const float* __restrict__ bih,
                                                      float* __restrict__ gi) {
  __shared__ float xs[32 * HSTRIDE];
  const int tid  = threadIdx.x;
  const int row0 = blockIdx.x * 32;
  for (int i = tid; i < 32 * D_; i += 512) {
    int r = i >> 9, c = i & (D_ - 1);
    xs[r * HSTRIDE + c] = x[(size_t)(row0 + r) * D_ + c];
  }
  __syncthreads();
  const int lane = tid & 31;
  const int wave = tid >> 5;
  const int w8   = wave & 7;
  const int msub = (wave >> 3) << 4;   // 0 or 16
  const int half = lane >> 4;
  const int l16  = lane & 15;
  const int kh   = half << 1;
  int nIdx[12];
#pragma unroll
  for (int j = 0; j < 12; ++j) nIdx[j] = (w8 + 8 * j) * 16 + l16;
  v8f acc[12];
#pragma unroll
  for (int j = 0; j < 12; ++j) acc[j] = 0.f;
  const int arow = (msub + l16) * HSTRIDE + kh;
  for (int k0 = 0; k0 < D_; k0 += 4) {
    v2f a = *(const v2f*)&xs[arow + k0];
    const float* wp = WihP + (k0 >> 2) * (G3 * 4) + kh;
#pragma unroll
    for (int j = 0; j < 12; ++j) {
      v2f b = *(const v2f*)&wp[nIdx[j] * 4];
      acc[j] = wmma4(a, b, acc[j]);
    }
  }
#pragma unroll
  for (int j = 0; j < 12; ++j) {
    const int nn = nIdx[j];
    const float bias = bih[nn];
#pragma unroll
    for (int g = 0; g < 8; ++g) {
      int m = row0 + msub + g + (half << 3);
      gi[(size_t)m * G3 + nn] = acc[j][g] + bias;
    }
  }
}

// ---------------------------------------------------------------------------
// Kernel 5: 12-step GRU. h tile (32 seq x 512) lives in LDS.
// Wave w owns feature tiles {w, w+8, w+16, w+24} for ALL three gates, so the
// gate fusion is in-wave. gi rows gathered via widx (pad row = b_ih).
// ---------------------------------------------------------------------------
__global__ __launch_bounds__(512) void gru_kernel(const float* __restrict__ gi,
                                                  const float* __restrict__ WhhP,
                                                  const float* __restrict__ bhh,
                                                  const int* __restrict__ widx,
                                                  float* __restrict__ out) {
  __shared__ float hs[32 * HSTRIDE];
  __shared__ int wl[32 * WIN];
  const int tid = threadIdx.x;
  const int m0  = blockIdx.x * 32;
  for (int i = tid; i < 32 * HSTRIDE; i += 512) hs[i] = 0.f;        // h0 = 0
  for (int i = tid; i < 32 * WIN; i += 512) wl[i] = widx[(size_t)m0 * WIN + i];
  const int lane = tid & 31;
  const int wave = tid >> 5;
  const int w8   = wave & 7;
  const int msub = (wave >> 3) << 4;
  const int half = lane >> 4;
  const int l16  = lane & 15;
  const int kh   = half << 1;
  int nIdx[12];      // [g*4 + fi]
  float bh[12];
#pragma unroll
  for (int g = 0; g < 3; ++g)
#pragma unroll
    for (int fi = 0; fi < 4; ++fi) {
      int f = (w8 + 8 * fi) * 16 + l16;
      nIdx[g * 4 + fi] = g * D_ + f;
      bh[g * 4 + fi]   = bhh[g * D_ + f];
    }
  const int arow = (msub + l16) * HSTRIDE + kh;

  for (int step = 0; step < WIN; ++step) {
    __syncthreads();                       // hs ready for everyone
    v8f acc[12];
#pragma unroll
    for (int j = 0; j < 12; ++j) acc[j] = 0.f;
    for (int k0 = 0; k0 < D_; k0 += 4) {   // gh = h @ W_hh^T
      v2f a = *(const v2f*)&hs[arow + k0];
      const float* wp = WhhP + (k0 >> 2) * (G3 * 4) + kh;
#pragma unroll
      for (int j = 0; j < 12; ++j) {
        v2f b = *(const v2f*)&wp[nIdx[j] * 4];
        acc[j] = wmma4(a, b, acc[j]);
      }
    }
    __syncthreads();                       // all reads of hs done
#pragma unroll
    for (int fi = 0; fi < 4; ++fi) {
      const int f = (w8 + 8 * fi) * 16 + l16;
#pragma unroll
      for (int g = 0; g < 8; ++g) {
        const int m   = msub + g + (half << 3);
        const int row = wl[m * WIN + step];
        const float* gr = gi + (size_t)row * G3;
        float r  = sigmoidf_(gr[f]          + acc[0 + fi][g] + bh[0 + fi]);
        float z  = sigmoidf_(gr[D_ + f]     + acc[4 + fi][g] + bh[4 + fi]);
        float nn = tanhf(gr[2 * D_ + f] + r * (acc[8 + fi][g] + bh[8 + fi]));
        float ho = hs[m * HSTRIDE + f];
        hs[m * HSTRIDE + f] = (1.f - z) * nn + z * ho;   // each (m,f) owned by one lane
      }
    }
  }
  __syncthreads();
  for (int i = tid; i < 32 * D_; i += 512) {
    int r = i >> 9, c = i & (D_ - 1);
    out[(size_t)(m0 + r) * D_ + c] = hs[r * HSTRIDE + c];
  }
}

// ---------------------------------------------------------------------------
extern "C" void kernel_launch(void* const* d_in, const int* in_sizes, int n_in,
                              void* d_out, int out_size, void* d_ws, size_t ws_size,
                              hipStream_t stream) {
  const float* x   = (const float*)d_in[0];   // [16,512,512]
  const float* Wih = (const float*)d_in[1];   // [1536,512]
  const float* Whh = (const float*)d_in[2];   // [1536,512]
  const float* bih = (const float*)d_in[3];   // [1536]
  const float* bhh = (const float*)d_in[4];   // [1536]
  float* out = (float*)d_out;                 // h [8192*512] ++ feat_len [16]

  char* ws = (char*)d_ws;
  // ws layout (bytes): ids 384K | widx 384K | WihP 3M | WhhP 3M | gi 48M  (~57.4MB total)
  int*   ids  = (int*)(ws + 0);
  int*   widx = (int*)(ws + 393216);
  float* WihP = (float*)(ws + 786432);
  float* WhhP = (float*)(ws + 786432 + 3145728);
  float* gi   = (float*)(ws + 786432 + 2 * 3145728);   // 8193 rows x 1536

  {
    int total = 2 * G3 * D_ + G3;
    prep_kernel<<<(total + 255) / 256, 256, 0, stream>>>(Wih, Whh, bih, WihP, WhhP, gi);
  }
  topk_kernel<<<NSEQ, 32, 0, stream>>>(x, ids);
  dedup_kernel<<<B_, T_, 0, stream>>>(ids, widx, out + (size_t)NSEQ * D_);
  gi_gemm_kernel<<<NSEQ / 32, 512, 0, stream>>>(x, WihP, bih, gi);
  gru_kernel<<<NSEQ / 32, 512, 0, stream>>>(gi, WhhP, bhh, widx, out);
}